// Gemma4TextRouter_87213605912609
// MI455X (gfx1250) — compile-verified
//
#include <hip/hip_runtime.h>
#include <math.h>

typedef __attribute__((ext_vector_type(2))) float v2f;
typedef __attribute__((ext_vector_type(8))) float v8f;

#define HIDDEN 4096
#define NEXP 64
#define TOPK 2
#define EPSV 1e-6f
#define TOKENS_PER_WAVE 16
#define WAVES_PER_BLOCK 8
#define TOKENS_PER_BLOCK (TOKENS_PER_WAVE * WAVES_PER_BLOCK)

// ---------------------------------------------------------------------------
// Kernel 1: fold elementwise scale and HIDDEN^-0.5 (=1/64) into router weight.
// W' is 64x4096 fp32 = 1 MB -> lives in L2 and is reused by every wave.
// ---------------------------------------------------------------------------
__global__ void prep_weights(const float* __restrict__ W,
                             const float* __restrict__ scale,
                             float* __restrict__ Wp) {
    int i = blockIdx.x * blockDim.x + threadIdx.x;
    if (i < NEXP * HIDDEN) {
        int h = i & (HIDDEN - 1);
        Wp[i] = W[i] * scale[h] * 0.015625f; // 4096^-0.5 == 1/64
    }
}

__device__ __forceinline__ v8f wmma_f32(v2f a, v2f b, v8f c) {
    // V_WMMA_F32_16X16X4_F32 : D(16x16 f32) = A(16x4 f32) x B(4x16 f32) + C
    return __builtin_amdgcn_wmma_f32_16x16x4_f32(
        /*neg_a=*/false, a, /*neg_b=*/false, b,
        /*c_mod=*/(short)0, c, /*reuse_a=*/false, /*reuse_b=*/false);
}

__device__ __forceinline__ bool better(float va, int ia, float vb, int ib) {
    // strict ordering with lower-index tie-break (matches lax.top_k)
    return (va > vb) || (va == vb && ia < ib);
}

// ---------------------------------------------------------------------------
// Kernel 2: fused single-pass router.
// Per wave: 16 tokens. K-loop streams x once (NT) computing sum(x^2) AND the
// raw score matmul S = x @ W'^T with V_WMMA_F32_16X16X4_F32 (4 expert tiles).
// Then scores = S * rsqrt(mean(x^2)+eps), softmax over 64, top-2.
// ---------------------------------------------------------------------------
__global__ void __launch_bounds__(256)
router_kernel(const float* __restrict__ x,
              const float* __restrict__ Wp,
              const float* __restrict__ pes,       // per_expert_scale [64]
              float* __restrict__ probs_out,       // [tokens, 64]
              float* __restrict__ w_out,           // [tokens, 2]
              int*   __restrict__ i_out) {         // [tokens, 2]
    const int lane = threadIdx.x & 31;
    const int wave = threadIdx.x >> 5;
    const int half = lane >> 4;   // 0: K+{0,1}, 1: K+{2,3} (A frag halves)
    const int lid  = lane & 15;

    const int token_base = blockIdx.x * TOKENS_PER_BLOCK + wave * TOKENS_PER_WAVE;

    // A fragment source: token row (token_base + lid), K offset half*2
    const float* xrow = x + (size_t)(token_base + lid) * HIDDEN + (half << 1);
    // B fragment sources: expert columns et*16 + lid, same K addressing
    const float* w0 = Wp + (size_t)(0 * 16 + lid) * HIDDEN + (half << 1);
    const float* w1 = Wp + (size_t)(1 * 16 + lid) * HIDDEN + (half << 1);
    const float* w2 = Wp + (size_t)(2 * 16 + lid) * HIDDEN + (half << 1);
    const float* w3 = Wp + (size_t)(3 * 16 + lid) * HIDDEN + (half << 1);

    v8f c0 = {}, c1 = {}, c2 = {}, c3 = {};
    float sumsq = 0.0f;

#pragma unroll 4
    for (int k = 0; k < HIDDEN; k += 4) {
        // x is streamed exactly once -> non-temporal so W' stays L2-resident
        v2f a = __builtin_nontemporal_load((const v2f*)(xrow + k));
        sumsq = fmaf(a.x, a.x, sumsq);
        sumsq = fmaf(a.y, a.y, sumsq);
        v2f b0 = *(const v2f*)(w0 + k);
        v2f b1 = *(const v2f*)(w1 + k);
        v2f b2 = *(const v2f*)(w2 + k);
        v2f b3 = *(const v2f*)(w3 + k);
        c0 = wmma_f32(a, b0, c0);
        c1 = wmma_f32(a, b1, c1);
        c2 = wmma_f32(a, b2, c2);
        c3 = wmma_f32(a, b3, c3);
    }

    // lanes l and l+16 each hold half the squares of token (l&15): combine.
    sumsq += __shfl_xor(sumsq, 16, 32);
    // rms reciprocal for token (lane & 15), replicated on both halves
    const float r = rsqrtf(sumsq * (1.0f / (float)HIDDEN) + EPSV);

#pragma unroll
    for (int v = 0; v < 8; ++v) {
        // C/D layout: VGPR v, lanes 0-15 -> M=v ; lanes 16-31 -> M=v+8
        const int t = token_base + (half << 3) + v;
        // fetch r for token M from the lane that owns it
        const float rv = __shfl(r, (half << 3) + v, 32);

        float s0 = c0[v] * rv;
        float s1 = c1[v] * rv;
        float s2 = c2[v] * rv;
        float s3 = c3[v] * rv;

        // softmax over 64 = 4 per-lane values x 16 lanes of this half-wave.
        float m = fmaxf(fmaxf(s0, s1), fmaxf(s2, s3));
#pragma unroll
        for (int off = 1; off < 16; off <<= 1)
            m = fmaxf(m, __shfl_xor(m, off, 32));

        float e0 = expf(s0 - m);
        float e1 = expf(s1 - m);
        float e2 = expf(s2 - m);
        float e3 = expf(s3 - m);
        float sum = e0 + e1 + e2 + e3;
#pragma unroll
        for (int off = 1; off < 16; off <<= 1)
            sum += __shfl_xor(sum, off, 32);

        const float inv = 1.0f / sum;
        const float p0 = e0 * inv, p1 = e1 * inv, p2 = e2 * inv, p3 = e3 * inv;

        // router_probabilities: lane lid writes experts {lid, 16+lid, 32+lid, 48+lid}
        float* pr = probs_out + (size_t)t * NEXP + lid;
        pr[0]  = p0;
        pr[16] = p1;
        pr[32] = p2;
        pr[48] = p3;

        // per-lane top-2 among its 4 experts (ascending index -> tie keeps lower)
        float b1v = -INFINITY, b2v = -INFINITY;
        int   b1i = 0x7fffffff, b2i = 0x7fffffff;
        const float pv[4] = {p0, p1, p2, p3};
#pragma unroll
        for (int et = 0; et < 4; ++et) {
            const float pe = pv[et];
            const int   ie = et * 16 + lid;
            if (better(pe, ie, b1v, b1i)) {
                b2v = b1v; b2i = b1i; b1v = pe; b1i = ie;
            } else if (better(pe, ie, b2v, b2i)) {
                b2v = pe; b2i = ie;
            }
        }
        // cross-lane top-2 merge within the 16-lane half
#pragma unroll
        for (int off = 1; off < 16; off <<= 1) {
            const float o1v = __shfl_xor(b1v, off, 32);
            const float o2v = __shfl_xor(b2v, off, 32);
            const int   o1i = __shfl_xor(b1i, off, 32);
            const int   o2i = __shfl_xor(b2i, off, 32);
            if (better(o1v, o1i, b1v, b1i)) {
                float n2v; int n2i;
                if (better(b1v, b1i, o2v, o2i)) { n2v = b1v; n2i = b1i; }
                else                            { n2v = o2v; n2i = o2i; }
                b1v = o1v; b1i = o1i; b2v = n2v; b2i = n2i;
            } else if (better(o1v, o1i, b2v, b2i)) {
                b2v = o1v; b2i = o1i;
            }
        }

        if (lid == 0) {
            const float wsum = b1v + b2v;
            const float iw = 1.0f / wsum;
            w_out[(size_t)t * 2 + 0] = b1v * iw * pes[b1i];
            w_out[(size_t)t * 2 + 1] = b2v * iw * pes[b2i];
            i_out[(size_t)t * 2 + 0] = b1i;
            i_out[(size_t)t * 2 + 1] = b2i;
        }
    }
}

extern "C" void kernel_launch(void* const* d_in, const int* in_sizes, int n_in,
                              void* d_out, int out_size, void* d_ws, size_t ws_size,
                              hipStream_t stream) {
    const float* hidden = (const float*)d_in[0];   // [4,4096,4096] f32
    const float* projw  = (const float*)d_in[1];   // [64,4096] f32
    const float* scale  = (const float*)d_in[2];   // [4096] f32
    const float* pes    = (const float*)d_in[3];   // [64] f32

    const int tokens = in_sizes[0] / HIDDEN;       // 16384

    float* Wp = (float*)d_ws;                      // 64*4096 f32 = 1 MB scratch

    // outputs concatenated: probs [tokens,64] f32 | weights [tokens,2] f32 | idx [tokens,2] i32
    float* probs = (float*)d_out;
    float* wout  = probs + (size_t)tokens * NEXP;
    int*   iout  = (int*)(wout + (size_t)tokens * TOPK);

    prep_weights<<<(NEXP * HIDDEN + 255) / 256, 256, 0, stream>>>(projw, scale, Wp);
    router_kernel<<<tokens / TOKENS_PER_BLOCK, 256, 0, stream>>>(
        hidden, Wp, pes, probs, wout, iout);
}